// GCN_65274912964675
// MI455X (gfx1250) — compile-verified
//
#include <hip/hip_runtime.h>
#include <hip/hip_bf16.h>

typedef __attribute__((ext_vector_type(2))) float v2f;
typedef __attribute__((ext_vector_type(8))) float v8f;

#define D_FEAT 128

// ---------------------------------------------------------------------------
// deg[i] = 1.0 (implicit self loop)
__global__ __launch_bounds__(256) void gcn_init_deg(float* __restrict__ deg, int n) {
    int i = blockIdx.x * 256 + threadIdx.x;
    if (i < n) deg[i] = 1.0f;
}

// deg[dst[e]] += 1 for every edge
__global__ __launch_bounds__(256) void gcn_deg_accum(const long long* __restrict__ dst,
                                                     float* __restrict__ deg,
                                                     long long nEdges) {
    long long e = (long long)blockIdx.x * 256 + threadIdx.x;
    if (e < nEdges) atomicAdd(&deg[(int)dst[e]], 1.0f);
}

// deg -> rsqrt(deg) in place
__global__ __launch_bounds__(256) void gcn_rsqrt(float* __restrict__ deg, int n) {
    int i = blockIdx.x * 256 + threadIdx.x;
    if (i < n) deg[i] = rsqrtf(deg[i]);
}

// zero a float buffer (grid-stride)
__global__ __launch_bounds__(256) void gcn_zero(float* __restrict__ p, size_t n) {
    size_t i = (size_t)blockIdx.x * 256 + threadIdx.x;
    size_t stride = (size_t)gridDim.x * 256;
    for (; i < n; i += stride) p[i] = 0.0f;
}

// ---------------------------------------------------------------------------
// H = X @ W   (N x 128) @ (128 x 128), fp32 WMMA 16x16x4.
// One wave computes one 16x16 output tile; 8 waves/block cover the 8 N-tiles
// of a 16-row slab so A-row loads are shared through L1.
__global__ __launch_bounds__(256) void gcn_gemm_wmma(const float* __restrict__ X,
                                                     const float* __restrict__ W,
                                                     float* __restrict__ H) {
    const int lane  = threadIdx.x & 31;
    const int wave  = threadIdx.x >> 5;      // 0..7 -> N-tile
    const int tileM = blockIdx.x * 16;       // N_NODES % 16 == 0
    const int tileN = wave * 16;
    const int half  = lane >> 4;             // 0: K pair {0,1}, 1: K pair {2,3}
    const int l15   = lane & 15;

    const float* __restrict__ xrow = X + (size_t)(tileM + l15) * D_FEAT;

    v8f c = {0.f, 0.f, 0.f, 0.f, 0.f, 0.f, 0.f, 0.f};

    #pragma unroll
    for (int k = 0; k < D_FEAT; k += 4) {
        const int k0 = k + 2 * half;
        // A 16x4: lane(0-15)=row M, VGPR0/1 = K0/K1 ; lanes 16-31 = K2/K3
        v2f a;
        a.x = xrow[k0];
        a.y = xrow[k0 + 1];
        // B 4x16: lane(0-15)=col N, VGPR0/1 = K0/K1 ; lanes 16-31 = K2/K3
        v2f b;
        b.x = W[(size_t)(k0)     * D_FEAT + tileN + l15];
        b.y = W[(size_t)(k0 + 1) * D_FEAT + tileN + l15];
        // 8 args: (neg_a, A, neg_b, B, c_mod, C, reuse_a, reuse_b)
        c = __builtin_amdgcn_wmma_f32_16x16x4_f32(false, a, false, b,
                                                  (short)0, c, false, false);
    }

    // C/D 16x16 fp32: VGPR v holds row M = v + 8*half, col N = l15
    #pragma unroll
    for (int v = 0; v < 8; ++v) {
        const int m = tileM + v + 8 * half;
        H[(size_t)m * D_FEAT + tileN + l15] = c[v];
    }
}

// ---------------------------------------------------------------------------
// One wave per edge: AGG[dst] += H[src] * (dinv[src]*dinv[dst])
__global__ __launch_bounds__(256) void gcn_aggregate(const float* __restrict__ H,
                                                     const long long* __restrict__ src,
                                                     const long long* __restrict__ dst,
                                                     const float* __restrict__ dinv,
                                                     float* __restrict__ AGG,
                                                     long long nEdges) {
    const long long e = (long long)blockIdx.x * 8 + (threadIdx.x >> 5);
    if (e >= nEdges) return;
    const int lane = threadIdx.x & 31;
    const int s = (int)src[e];
    const int d = (int)dst[e];
    const float coef = dinv[s] * dinv[d];

    const float4 hv = ((const float4*)(H + (size_t)s * D_FEAT))[lane];
    float* __restrict__ ad = AGG + (size_t)d * D_FEAT + lane * 4;
    atomicAdd(ad + 0, hv.x * coef);
    atomicAdd(ad + 1, hv.y * coef);
    atomicAdd(ad + 2, hv.z * coef);
    atomicAdd(ad + 3, hv.w * coef);
}

// ---------------------------------------------------------------------------
// O[i][f] = relu(AGG[i][f] + H[i][f]*dinv[i]^2 + bias[f])   (in-place on AGG ok)
__global__ __launch_bounds__(128) void gcn_finalize(float* __restrict__ O,
                                                    const float* __restrict__ AGG,
                                                    const float* __restrict__ H,
                                                    const float* __restrict__ dinv,
                                                    const float* __restrict__ bias) {
    const int i = blockIdx.x;
    const int f = threadIdx.x;
    const float di = dinv[i];
    const size_t idx = (size_t)i * D_FEAT + f;
    float v = AGG[idx] + H[idx] * (di * di) + bias[f];
    O[idx] = v > 0.0f ? v : 0.0f;
}

// ---------------------------------------------------------------------------
extern "C" void kernel_launch(void* const* d_in, const int* in_sizes, int n_in,
                              void* d_out, int out_size, void* d_ws, size_t ws_size,
                              hipStream_t stream) {
    const float*     x   = (const float*)d_in[0];
    const long long* ei  = (const long long*)d_in[1];   // int64 [2, E]
    const float*     W1  = (const float*)d_in[2];
    const float*     b1  = (const float*)d_in[3];
    const float*     W2  = (const float*)d_in[4];
    const float*     b2  = (const float*)d_in[5];
    float*           out = (float*)d_out;

    const int       nNodes = in_sizes[0] / D_FEAT;      // 100000 (multiple of 16)
    const long long nEdges = in_sizes[1] / 2;           // 1600000

    const long long* src = ei;
    const long long* dst = ei + nEdges;

    // workspace layout: dinv | H | AGG
    float* deg = (float*)d_ws;                          // nNodes floats -> dinv
    float* h   = deg + ((nNodes + 63) & ~63);           // nNodes*128
    float* agg = h + (size_t)nNodes * D_FEAT;           // nNodes*128

    const size_t feat = (size_t)nNodes * D_FEAT;
    const int gridN     = (nNodes + 255) / 256;
    const int gridE     = (int)((nEdges + 255) / 256);
    const int gridEdge  = (int)((nEdges + 7) / 8);      // 1 wave / edge
    const int gridGemm  = nNodes / 16;                  // 6250
    const int gridZero  = 2048;

    // symmetric-norm coefficients
    gcn_init_deg <<<gridN, 256, 0, stream>>>(deg, nNodes);
    gcn_deg_accum<<<gridE, 256, 0, stream>>>(dst, deg, nEdges);
    gcn_rsqrt    <<<gridN, 256, 0, stream>>>(deg, nNodes);

    // ---- layer 1: relu(gcn(x, W1, b1)) -> agg ----
    gcn_gemm_wmma<<<gridGemm, 256, 0, stream>>>(x, W1, h);
    gcn_zero     <<<gridZero, 256, 0, stream>>>(agg, feat);
    gcn_aggregate<<<gridEdge, 256, 0, stream>>>(h, src, dst, deg, agg, nEdges);
    gcn_finalize <<<nNodes, 128, 0, stream>>>(agg, agg, h, deg, b1);

    // ---- layer 2: relu(gcn(agg, W2, b2)) -> out ----
    gcn_gemm_wmma<<<gridGemm, 256, 0, stream>>>(agg, W2, h);
    gcn_zero     <<<gridZero, 256, 0, stream>>>(out, feat);
    gcn_aggregate<<<gridEdge, 256, 0, stream>>>(h, src, dst, deg, out, nEdges);
    gcn_finalize <<<nNodes, 128, 0, stream>>>(out, out, h, deg, b2);
}